// FeedbackModel_24592982737431
// MI455X (gfx1250) — compile-verified
//
#include <hip/hip_runtime.h>

#define N_NODES  100000
#define N_EDGES  600000
#define EMB_DIM  300
#define F1       128
#define F2       64
#define N_GRAPHS 256
#define N_OUT    6

typedef float v2f __attribute__((ext_vector_type(2)));
typedef float v8f __attribute__((ext_vector_type(8)));

__device__ __forceinline__ void atomicAddF(float* p, float v) {
    (void)__hip_atomic_fetch_add(p, v, __ATOMIC_RELAXED, __HIP_MEMORY_SCOPE_AGENT);
}

// ---------------- degree / norm ----------------
__global__ void k_init_deg(float* __restrict__ deg) {
    int i = blockIdx.x * blockDim.x + threadIdx.x;
    if (i < N_NODES) deg[i] = 1.0f;            // self-loop contributes 1
}

__global__ void k_deg_edges(const int* __restrict__ col, float* __restrict__ deg) {
    int e = blockIdx.x * blockDim.x + threadIdx.x;
    if (e < N_EDGES) atomicAddF(&deg[col[e]], 1.0f);
}

__global__ void k_rsqrt(float* __restrict__ d) {
    int i = blockIdx.x * blockDim.x + threadIdx.x;
    if (i < N_NODES) d[i] = rsqrtf(d[i]);      // deg >= 1 always (self-loops)
}

// ---------------- GEMM1: h1 = emb[x] @ W1  (fused gather, fp32 WMMA) ----------------
// Block = 256 threads = 8 waves; 16 nodes per block; wave w owns N-tile w (16 cols of 128).
__launch_bounds__(256)
__global__ void k_gemm1(const int* __restrict__ x, const float* __restrict__ emb,
                        const float* __restrict__ W1, float* __restrict__ h1) {
    __shared__ int   xs[16];
    __shared__ float As[16 * 301];             // odd stride -> conflict-free column reads
    const int tid  = threadIdx.x;
    const int base = blockIdx.x * 16;          // N_NODES % 16 == 0

    if (tid < 16) xs[tid] = x[base + tid];
    __syncthreads();
    for (int idx = tid; idx < 16 * EMB_DIM; idx += 256) {
        int r = idx / EMB_DIM, c = idx - r * EMB_DIM;
        As[r * 301 + c] = emb[(size_t)xs[r] * EMB_DIM + c];
    }
    __syncthreads();

    const int lane = tid & 31;
    const int wave = tid >> 5;                 // ntile 0..7
    const int m    = lane & 15;
    const int koff = (lane >> 4) << 1;         // lanes 0-15: K=0,1 ; lanes 16-31: K=2,3
    const int ncol = wave * 16 + m;

    v8f acc = {};
    for (int k = 0; k < EMB_DIM; k += 4) {     // 300 % 4 == 0
        v2f a, b;
        a.x = As[m * 301 + k + koff];
        a.y = As[m * 301 + k + koff + 1];
        b.x = W1[(k + koff)     * F1 + ncol];
        b.y = W1[(k + koff + 1) * F1 + ncol];
        acc = __builtin_amdgcn_wmma_f32_16x16x4_f32(false, a, false, b,
                                                    (short)0, acc, false, false);
    }
    const int mrow = base + (lane >> 4) * 8;   // lanes 0-15: M=j ; lanes 16-31: M=j+8
#pragma unroll
    for (int j = 0; j < 8; ++j)
        h1[(size_t)(mrow + j) * F1 + ncol] = acc[j];
}

// ---------------- self-loop init: agg = h * dinv^2 (non-atomic full overwrite) ------
template <int F>
__global__ void k_selfloop(const float* __restrict__ h, const float* __restrict__ dinv,
                           float* __restrict__ agg) {
    int i = blockIdx.x * blockDim.x + threadIdx.x;
    if (i < N_NODES * F) {
        int n = i / F;                          // F power of 2 -> shift
        float d = dinv[n];
        agg[i] = h[i] * d * d;
    }
}

// ---------------- edge scatter, conv1 (F=128): one wave per edge, lane -> float4 ----
__launch_bounds__(256)
__global__ void k_edges1(const int* __restrict__ row, const int* __restrict__ col,
                         const float* __restrict__ dinv, const float* __restrict__ h,
                         float* __restrict__ agg) {
    int e = (blockIdx.x * 256 + threadIdx.x) >> 5;
    if (e >= N_EDGES) return;
    int lane = threadIdx.x & 31;
    int r = row[e], c = col[e];
    float nrm = dinv[r] * dinv[c];
    const float4 v = *reinterpret_cast<const float4*>(&h[(size_t)r * F1 + lane * 4]);
    float* dst = &agg[(size_t)c * F1 + lane * 4];
    atomicAddF(dst + 0, v.x * nrm);
    atomicAddF(dst + 1, v.y * nrm);
    atomicAddF(dst + 2, v.z * nrm);
    atomicAddF(dst + 3, v.w * nrm);
}

// ---------------- edge scatter, conv2 (F=64): one wave per edge, lane -> float2 -----
__launch_bounds__(256)
__global__ void k_edges2(const int* __restrict__ row, const int* __restrict__ col,
                         const float* __restrict__ dinv, const float* __restrict__ h,
                         float* __restrict__ agg) {
    int e = (blockIdx.x * 256 + threadIdx.x) >> 5;
    if (e >= N_EDGES) return;
    int lane = threadIdx.x & 31;
    int r = row[e], c = col[e];
    float nrm = dinv[r] * dinv[c];
    const float2 v = *reinterpret_cast<const float2*>(&h[(size_t)r * F2 + lane * 2]);
    float* dst = &agg[(size_t)c * F2 + lane * 2];
    atomicAddF(dst + 0, v.x * nrm);
    atomicAddF(dst + 1, v.y * nrm);
}

// ---------------- GEMM2: h2 = relu(h1_agg + b1) @ W2  (bias+relu fused into A load) -
// Block = 256 threads = 8 waves; 32 nodes (2 M-tiles) x 64 cols (4 N-tiles).
__launch_bounds__(256)
__global__ void k_gemm2(const float* __restrict__ h1agg, const float* __restrict__ b1,
                        const float* __restrict__ W2, float* __restrict__ h2) {
    __shared__ float As[32 * 129];
    const int tid  = threadIdx.x;
    const int base = blockIdx.x * 32;          // N_NODES % 32 == 0

    for (int idx = tid; idx < 32 * F1; idx += 256) {
        int r = idx >> 7, c = idx & 127;
        float v = h1agg[(size_t)(base + r) * F1 + c] + b1[c];
        As[r * 129 + c] = fmaxf(v, 0.0f);
    }
    __syncthreads();

    const int lane = tid & 31;
    const int wave = tid >> 5;
    const int mt   = wave >> 2;                // 0..1
    const int nt   = wave & 3;                 // 0..3
    const int m    = lane & 15;
    const int koff = (lane >> 4) << 1;
    const int ncol = nt * 16 + m;

    v8f acc = {};
    for (int k = 0; k < F1; k += 4) {
        v2f a, b;
        a.x = As[(mt * 16 + m) * 129 + k + koff];
        a.y = As[(mt * 16 + m) * 129 + k + koff + 1];
        b.x = W2[(k + koff)     * F2 + ncol];
        b.y = W2[(k + koff + 1) * F2 + ncol];
        acc = __builtin_amdgcn_wmma_f32_16x16x4_f32(false, a, false, b,
                                                    (short)0, acc, false, false);
    }
    const int mrow = base + mt * 16 + (lane >> 4) * 8;
#pragma unroll
    for (int j = 0; j < 8; ++j)
        h2[(size_t)(mrow + j) * F2 + ncol] = acc[j];
}

// ---------------- pooling ----------------
__global__ void k_zero(float* __restrict__ p, int n) {
    int i = blockIdx.x * blockDim.x + threadIdx.x;
    if (i < n) p[i] = 0.0f;
}

// One wave per node; lane handles 2 features. relu(h2_agg + b2) fused here.
__launch_bounds__(256)
__global__ void k_pool(const int* __restrict__ batch, const float* __restrict__ h2agg,
                       const float* __restrict__ b2, float* __restrict__ sums,
                       float* __restrict__ cnt) {
    int n = (blockIdx.x * 256 + threadIdx.x) >> 5;
    if (n >= N_NODES) return;
    int lane = threadIdx.x & 31;
    int g = batch[n];
    int f = lane * 2;
    float v0 = fmaxf(h2agg[(size_t)n * F2 + f    ] + b2[f    ], 0.0f);
    float v1 = fmaxf(h2agg[(size_t)n * F2 + f + 1] + b2[f + 1], 0.0f);
    atomicAddF(&sums[g * F2 + f    ], v0);
    atomicAddF(&sums[g * F2 + f + 1], v1);
    if (lane == 0) atomicAddF(&cnt[g], 1.0f);
}

// out[g][o] = relu( (sums[g]/max(cnt,1)) . Wout[:,o] + bout[o] )
__global__ void k_head(const float* __restrict__ sums, const float* __restrict__ cnt,
                       const float* __restrict__ Wout, const float* __restrict__ bout,
                       float* __restrict__ out) {
    int g = threadIdx.x;                       // 0..255
    int o = blockIdx.x;                        // 0..5
    float denom = fmaxf(cnt[g], 1.0f);
    float acc = 0.0f;
    for (int f = 0; f < F2; ++f)
        acc += sums[g * F2 + f] * Wout[f * N_OUT + o];
    out[g * N_OUT + o] = fmaxf(acc / denom + bout[o], 0.0f);
}

// ---------------- launch ----------------
extern "C" void kernel_launch(void* const* d_in, const int* in_sizes, int n_in,
                              void* d_out, int out_size, void* d_ws, size_t ws_size,
                              hipStream_t stream) {
    const int*   x     = (const int*)  d_in[0];
    const int*   ei    = (const int*)  d_in[1];
    const int*   batch = (const int*)  d_in[2];
    const float* emb   = (const float*)d_in[3];
    const float* W1    = (const float*)d_in[4];
    const float* b1    = (const float*)d_in[5];
    const float* W2    = (const float*)d_in[6];
    const float* b2    = (const float*)d_in[7];
    const float* Wout  = (const float*)d_in[8];
    const float* bout  = (const float*)d_in[9];
    float* out = (float*)d_out;

    const int* row = ei;               // edge_index[0] = sources
    const int* col = ei + N_EDGES;     // edge_index[1] = targets

    // workspace carve-up (aliased: h2 reuses h1's space, h2_agg reuses h1_agg's)
    char* ws = (char*)d_ws;
    size_t off = 0;
    auto alloc = [&](size_t bytes) -> void* {
        void* p = ws + off;
        off = (off + bytes + 511) & ~(size_t)511;
        return p;
    };
    float* dinv = (float*)alloc((size_t)N_NODES * 4);
    float* bufA = (float*)alloc((size_t)N_NODES * F1 * 4);   // h1, later h2
    float* bufB = (float*)alloc((size_t)N_NODES * F1 * 4);   // h1_agg, later h2_agg
    float* sums = (float*)alloc((size_t)N_GRAPHS * F2 * 4);  // cnt directly follows
    float* cnt  = (float*)alloc((size_t)N_GRAPHS * 4);
    (void)cnt;
    cnt = sums + N_GRAPHS * F2;        // contiguous with sums (size is 512-aligned)

    float* h1    = bufA;
    float* h1agg = bufB;
    float* h2    = bufA;
    float* h2agg = bufB;

    // normalization
    k_init_deg <<<(N_NODES + 255) / 256, 256, 0, stream>>>(dinv);
    k_deg_edges<<<(N_EDGES + 255) / 256, 256, 0, stream>>>(col, dinv);
    k_rsqrt    <<<(N_NODES + 255) / 256, 256, 0, stream>>>(dinv);

    // conv1
    k_gemm1<<<N_NODES / 16, 256, 0, stream>>>(x, emb, W1, h1);
    k_selfloop<F1><<<(N_NODES * F1 + 255) / 256, 256, 0, stream>>>(h1, dinv, h1agg);
    k_edges1<<<(N_EDGES * 32 + 255) / 256, 256, 0, stream>>>(row, col, dinv, h1, h1agg);

    // conv2 (bias1+relu fused into GEMM2's LDS staging)
    k_gemm2<<<N_NODES / 32, 256, 0, stream>>>(h1agg, b1, W2, h2);
    k_selfloop<F2><<<(N_NODES * F2 + 255) / 256, 256, 0, stream>>>(h2, dinv, h2agg);
    k_edges2<<<(N_EDGES * 32 + 255) / 256, 256, 0, stream>>>(row, col, dinv, h2, h2agg);

    // pool + head (bias2+relu fused into pool)
    k_zero<<<(N_GRAPHS * F2 + N_GRAPHS + 255) / 256, 256, 0, stream>>>(sums, N_GRAPHS * F2 + N_GRAPHS);
    k_pool<<<(N_NODES * 32 + 255) / 256, 256, 0, stream>>>(batch, h2agg, b2, sums, cnt);
    k_head<<<N_OUT, N_GRAPHS, 0, stream>>>(sums, cnt, Wout, bout, out);
}